// NeuralFODE_69982197121563
// MI455X (gfx1250) — compile-verified
//
#include <hip/hip_runtime.h>
#include <math.h>

// NeuralFODE on MI455X (gfx1250, wave32).
//
// Key insight: the reference MLP has NO intermediate activations, so the
// 4-layer chain collapses to one affine map:
//   deriv = tanh(state @ Weff + t * w_t + beff)
// with Weff (64x64), w_t (64), beff (64) precomputed once.
// Per step FLOPs drop 172 GF -> 4.3 GF; the kernel becomes bound on the
// 128 MB trajectory write (~5.5us @ 23.3 TB/s).
//
// Kernel 1 (one WG, 1024 thr): collapse the weight chain into d_ws (fp32 VALU).
// Kernel 2 (32 WGs x 1 wave x 16 batch rows): sequential scan over T-1 steps,
// 64x64 fp32 matmul per step via V_WMMA_F32_16X16X4_F32 (full fp32 fidelity),
// Weff resident in 128 VGPRs, state double-homed in regs (C/D layout) and in
// LDS (stride 68 -> zero-bank-conflict A-fragment transposes each step).
//
// Round-1 fixes:
//  * tanh via native gfx1250 v_tanh_f32 (__builtin_amdgcn_tanhf) instead of
//    the branchy exp-based libm expansion on the serial critical path.
//  * no per-step barrier: single-wave block + in-order LDS pipeline make the
//    ds_store -> ds_load ordering a hardware guarantee; this stops
//    s_wait_storecnt 0 from serializing the HBM trajectory writes each step.

typedef float v2f __attribute__((ext_vector_type(2)));
typedef float v8f __attribute__((ext_vector_type(8)));

#define HID 256
#define RGAMMA 0.5641895835477563f   // 1 / Gamma(0.5) = 1/sqrt(pi)
#define SPAD 68                      // LDS row stride (words): 4*m+c covers all 64 banks

#if defined(__has_builtin)
#if __has_builtin(__builtin_amdgcn_tanhf)
#define FAST_TANH(x) __builtin_amdgcn_tanhf(x)
#endif
#endif
#ifndef FAST_TANH
#define FAST_TANH(x) tanhf(x)
#endif

// ---------------- kernel 1: collapse the affine chain ----------------
__global__ __launch_bounds__(1024)
void fode_collapse(const float* __restrict__ W0, const float* __restrict__ b0,
                   const float* __restrict__ W1, const float* __restrict__ b1,
                   const float* __restrict__ W2, const float* __restrict__ b2,
                   const float* __restrict__ W3, const float* __restrict__ b3,
                   float* __restrict__ ws) {
  float* Wfull = ws;                 // 65*64  (row 0 = w_t, rows 1..64 = Weff)
  float* beff  = ws + 65 * 64;       // 64
  float* tA    = ws + 4224;          // 65*256
  float* tB    = tA + 65 * HID;      // 65*256
  float* cb1   = tB + 65 * HID;      // 256
  float* cb2   = cb1 + HID;          // 256
  const int tid = threadIdx.x;

  // stage 1: tA = W0 @ W1   (65x256 = 65x256 @ 256x256), cb1 = b0@W1 + b1
  for (int idx = tid; idx < 65 * HID; idx += blockDim.x) {
    const int i = idx >> 8, j = idx & 255;
    float s = 0.f;
    for (int k = 0; k < HID; ++k) s = fmaf(W0[i * HID + k], W1[k * HID + j], s);
    tA[idx] = s;
  }
  if (tid < HID) {
    float s = b1[tid];
    for (int k = 0; k < HID; ++k) s = fmaf(b0[k], W1[k * HID + tid], s);
    cb1[tid] = s;
  }
  __syncthreads();

  // stage 2: tB = tA @ W2, cb2 = cb1@W2 + b2
  for (int idx = tid; idx < 65 * HID; idx += blockDim.x) {
    const int i = idx >> 8, j = idx & 255;
    float s = 0.f;
    for (int k = 0; k < HID; ++k) s = fmaf(tA[i * HID + k], W2[k * HID + j], s);
    tB[idx] = s;
  }
  __syncthreads();
  if (tid < HID) {
    float s = b2[tid];
    for (int k = 0; k < HID; ++k) s = fmaf(cb1[k], W2[k * HID + tid], s);
    cb2[tid] = s;
  }
  __syncthreads();

  // stage 3: Wfull = tB @ W3 (65x64), beff = cb2@W3 + b3
  for (int idx = tid; idx < 65 * 64; idx += blockDim.x) {
    const int i = idx >> 6, j = idx & 63;
    float s = 0.f;
    for (int k = 0; k < HID; ++k) s = fmaf(tB[i * HID + k], W3[k * 64 + j], s);
    Wfull[idx] = s;
  }
  if (tid < 64) {
    float s = b3[tid];
    for (int k = 0; k < HID; ++k) s = fmaf(cb2[k], W3[k * 64 + tid], s);
    beff[tid] = s;
  }
}

// ---------------- kernel 2: fractional-ODE scan ----------------
__global__ __launch_bounds__(32)
void fode_scan(const float* __restrict__ x, const float* __restrict__ tv,
               const float* __restrict__ ws, float* __restrict__ out,
               int T) {
  const float* Wfull = ws;
  const float* beff  = ws + 65 * 64;

  __shared__ float sst[16 * SPAD];   // state tile, padded: zero bank conflicts

  const int l   = threadIdx.x;       // lane 0..31 (wave32)
  const int hi  = l >> 4;            // half-wave
  const int lo  = l & 15;
  const int b0r = blockIdx.x * 16;   // this wave's 16 batch rows

  // --- persistent B fragments: Weff resident in 128 VGPRs ---
  // V_WMMA_F32_16X16X4_F32 B layout (4x16): lanes hold N, VGPR0/1 hold
  // K = 2*hi + {0,1}.  Row 0 of Wfull is the t-row -> skip with +1.
  v2f bf[16][4];
#pragma unroll
  for (int kk = 0; kk < 16; ++kk)
#pragma unroll
    for (int nt = 0; nt < 4; ++nt) {
      const int krow = 1 + 4 * kk + 2 * hi;
      const int col  = lo + 16 * nt;
      bf[kk][nt].x = Wfull[(krow + 0) * 64 + col];
      bf[kk][nt].y = Wfull[(krow + 1) * 64 + col];
    }
  float wt[4], bb[4];
#pragma unroll
  for (int nt = 0; nt < 4; ++nt) {
    wt[nt] = Wfull[lo + 16 * nt];    // t-row
    bb[nt] = beff[lo + 16 * nt];
  }

  // --- init state: registers in C/D layout + LDS copy; emit y(t=0) ---
  float st[4][8];
#pragma unroll
  for (int nt = 0; nt < 4; ++nt)
#pragma unroll
    for (int r = 0; r < 8; ++r) {
      const int m = r + 8 * hi, c = lo + 16 * nt;
      st[nt][r] = x[(b0r + m) * 64 + c];
    }
  for (int i = l; i < 16 * 64; i += 32) {
    const int m = i >> 6, k = i & 63;
    const float v = x[(b0r + m) * 64 + k];
    sst[m * SPAD + k] = v;
    out[(size_t)(b0r + m) * (size_t)T * 64 + k] = v;   // solution[:,0,:] = y0
  }
  __syncthreads();   // once, before the scan

  for (int step = 0; step < T - 1; ++step) {
    const float tp     = tv[step];
    const float dt     = tv[step + 1] - tp;
    const float factor = sqrtf(dt) * RGAMMA;           // dt^alpha / Gamma(alpha)

    // A fragments (16x4 fp32): lanes hold M, VGPR0/1 hold K = 2*hi + {0,1}.
    // LDS word index = 68*M + k -> 32 lanes touch 64 distinct banks: conflict-free.
    v2f a[16];
#pragma unroll
    for (int kk = 0; kk < 16; ++kk)
      a[kk] = *(const v2f*)&sst[lo * SPAD + 4 * kk + 2 * hi];

#pragma unroll
    for (int nt = 0; nt < 4; ++nt) {
      v8f acc = {0.f, 0.f, 0.f, 0.f, 0.f, 0.f, 0.f, 0.f};
#pragma unroll
      for (int kk = 0; kk < 16; ++kk)
        acc = __builtin_amdgcn_wmma_f32_16x16x4_f32(
            false, a[kk], false, bf[kk][nt], (short)0, acc, false, false);

      // epilogue: h = acc + t*w_t + beff ; state += factor * tanh(h)
#pragma unroll
      for (int r = 0; r < 8; ++r) {
        const float h = acc[r] + fmaf(tp, wt[nt], bb[nt]);
        const float s = fmaf(factor, FAST_TANH(h), st[nt][r]);
        st[nt][r] = s;
        const int m = r + 8 * hi, c = lo + 16 * nt;
        sst[m * SPAD + c] = s;                          // feed next step's A
        out[(size_t)(b0r + m) * (size_t)T * 64 +
            (size_t)(step + 1) * 64 + c] = s;           // trajectory write
      }
    }
    // no barrier: single-wave block; DS pipe is in-order within a wave, and
    // the compiler's s_wait_dscnt covers the load->use edge. Global stores
    // are free to pipeline across steps.
  }
}

extern "C" void kernel_launch(void* const* d_in, const int* in_sizes, int n_in,
                              void* d_out, int out_size, void* d_ws, size_t ws_size,
                              hipStream_t stream) {
  const float* x  = (const float*)d_in[0];
  const float* tv = (const float*)d_in[1];
  const float* W0 = (const float*)d_in[2];
  const float* b0 = (const float*)d_in[3];
  const float* W1 = (const float*)d_in[4];
  const float* b1 = (const float*)d_in[5];
  const float* W2 = (const float*)d_in[6];
  const float* b2 = (const float*)d_in[7];
  const float* W3 = (const float*)d_in[8];
  const float* b3 = (const float*)d_in[9];
  float* out = (float*)d_out;
  float* ws  = (float*)d_ws;

  const int B = in_sizes[0] / 64;   // 512
  const int T = in_sizes[1];        // 1024

  fode_collapse<<<1, 1024, 0, stream>>>(W0, b0, W1, b1, W2, b2, W3, b3, ws);
  fode_scan<<<B / 16, 32, 0, stream>>>(x, tv, ws, out, T);
}